// GATLayerMethod_86388972192204
// MI455X (gfx1250) — compile-verified
//
#include <hip/hip_runtime.h>
#include <hip/hip_bf16.h>

typedef float v2f __attribute__((ext_vector_type(2)));
typedef float v8f __attribute__((ext_vector_type(8)));

#define NODES      10000
#define DIM        128
#define EDGES      640000
#define NEG_SLOPE  0.2f
#define INV_SCALE  0.08838834764831845f   // 1/sqrt(128)

// ---------------------------------------------------------------------------
// Kernel 1: Y = X @ W^T with fp32 WMMA (16x16 tiles, K stepped by 4).
// mi=0 -> LQ (leaky), mi=1 -> LK (leaky), mi=2 -> V (plain).
// 625 M-tiles x 8 N-tiles x 3 matrices = 15000 waves, one tile per wave.
// ---------------------------------------------------------------------------
__global__ __launch_bounds__(128)
void qkv_wmma_kernel(const float* __restrict__ x,
                     const float* __restrict__ Wq,
                     const float* __restrict__ Wk,
                     const float* __restrict__ Wv,
                     float* __restrict__ LQ,
                     float* __restrict__ LK,
                     float* __restrict__ Vo)
{
    const int wavesPerBlk = blockDim.x >> 5;
    const int wave = blockIdx.x * wavesPerBlk + (threadIdx.x >> 5);
    const int lane = threadIdx.x & 31;

    const int M_TILES = NODES / 16;        // 625
    const int N_TILES = DIM / 16;          // 8
    const int TILES   = M_TILES * N_TILES; // 5000
    if (wave >= 3 * TILES) return;         // uniform per wave -> EXEC stays full

    const int mi = wave / TILES;
    const int t  = wave % TILES;
    const int mt = t / N_TILES;
    const int nt = t % N_TILES;

    const float* __restrict__ W = (mi == 0) ? Wq : (mi == 1) ? Wk : Wv;
    float* __restrict__ Y       = (mi == 0) ? LQ : (mi == 1) ? LK : Vo;

    const int m  = lane & 15;   // A: row in tile; B: out-column in tile
    const int hi = lane >> 4;   // lanes 16..31 carry K+2..K+3

    // A lane layout (16x4 f32): v[j] = X[m, k + 2*hi + j]
    const float* __restrict__ arow = x + (size_t)(mt * 16 + m) * DIM + 2 * hi;
    // B = W^T tile: B[k, o] = W[o, k]; v[j] = W[o, k + 2*hi + j] (contiguous in k)
    const float* __restrict__ brow = W + (size_t)(nt * 16 + m) * DIM + 2 * hi;

    v8f acc = {};
#pragma unroll
    for (int k = 0; k < DIM; k += 4) {
        v2f a = *(const v2f*)(arow + k);
        v2f b = *(const v2f*)(brow + k);
        acc = __builtin_amdgcn_wmma_f32_16x16x4_f32(
            /*neg_a=*/false, a, /*neg_b=*/false, b,
            /*c_mod=*/(short)0, acc, /*reuse_a=*/false, /*reuse_b=*/false);
    }

    // D layout: v[r] holds row (r + 8*hi), col = lane&15
    const bool leaky = (mi != 2);
#pragma unroll
    for (int r = 0; r < 8; ++r) {
        float v = acc[r];
        if (leaky) v = (v >= 0.f) ? v : NEG_SLOPE * v;
        Y[(size_t)(mt * 16 + r + 8 * hi) * DIM + nt * 16 + m] = v;
    }
}

// ---------------------------------------------------------------------------
// Kernel 2: zero output + den, init per-node max keys.
// ---------------------------------------------------------------------------
__global__ __launch_bounds__(256)
void init_kernel(float* __restrict__ out, float* __restrict__ den,
                 unsigned* __restrict__ maxkey)
{
    const int total = NODES * DIM;
    for (int i = blockIdx.x * blockDim.x + threadIdx.x; i < total;
         i += gridDim.x * blockDim.x) {
        out[i] = 0.f;
        if (i < NODES) { den[i] = 0.f; maxkey[i] = 0u; } // key 0 == "-inf"
    }
}

__device__ __forceinline__ unsigned f32_to_key(float f) {
    unsigned u = __float_as_uint(f);
    return (u & 0x80000000u) ? ~u : (u | 0x80000000u);
}
__device__ __forceinline__ float key_to_f32(unsigned k) {
    return (k & 0x80000000u) ? __uint_as_float(k & 0x7FFFFFFFu)
                             : __uint_as_float(~k);
}

// ---------------------------------------------------------------------------
// Kernel 3: one wave per edge -- 512B coalesced row reads, shuffle-reduce dot,
// atomicMax (uint monotone key) for segment max.
// ---------------------------------------------------------------------------
__global__ __launch_bounds__(256)
void edge_scores_kernel(const float* __restrict__ LQ,
                        const float* __restrict__ LK,
                        const long long* __restrict__ ei,
                        float* __restrict__ scores,
                        unsigned* __restrict__ maxkey)
{
    const int lane   = threadIdx.x & 31;
    const int wave   = (blockIdx.x * blockDim.x + threadIdx.x) >> 5;
    const int nwaves = (gridDim.x * blockDim.x) >> 5;

    for (int e = wave; e < EDGES; e += nwaves) {
        const int r = (int)ei[e];
        const int c = (int)ei[EDGES + e];
        float4 a = ((const float4*)(LQ + (size_t)r * DIM))[lane];
        float4 b = ((const float4*)(LK + (size_t)c * DIM))[lane];
        float v = a.x * b.x + a.y * b.y + a.z * b.z + a.w * b.w;
#pragma unroll
        for (int off = 16; off > 0; off >>= 1)
            v += __shfl_xor(v, off, 32);
        v *= INV_SCALE;
        if (lane == 0) {
            scores[e] = v;
            atomicMax(&maxkey[r], f32_to_key(v));
        }
    }
}

// ---------------------------------------------------------------------------
// Kernel 4: exp(s - mx[row]) in-place, atomicAdd into den[row].
// ---------------------------------------------------------------------------
__global__ __launch_bounds__(256)
void edge_exp_kernel(const long long* __restrict__ ei,
                     float* __restrict__ scores,          // in: s, out: exp
                     const unsigned* __restrict__ maxkey,
                     float* __restrict__ den)
{
    for (int e = blockIdx.x * blockDim.x + threadIdx.x; e < EDGES;
         e += gridDim.x * blockDim.x) {
        const int r = (int)ei[e];
        const float m  = key_to_f32(maxkey[r]);
        const float es = expf(scores[e] - m);
        scores[e] = es;
        atomicAdd(&den[r], es);
    }
}

// ---------------------------------------------------------------------------
// Kernel 5: one wave per edge; alpha * V[col] scattered into out[row] with
// native f32 global atomics.
// ---------------------------------------------------------------------------
__global__ __launch_bounds__(256)
void edge_accum_kernel(const long long* __restrict__ ei,
                       const float* __restrict__ expbuf,
                       const float* __restrict__ den,
                       const float* __restrict__ V,
                       float* __restrict__ out)
{
    const int lane   = threadIdx.x & 31;
    const int wave   = (blockIdx.x * blockDim.x + threadIdx.x) >> 5;
    const int nwaves = (gridDim.x * blockDim.x) >> 5;

    for (int e = wave; e < EDGES; e += nwaves) {
        const int r = (int)ei[e];
        const int c = (int)ei[EDGES + e];
        const float alpha = expbuf[e] / (den[r] + 1e-8f);
        float4 v = ((const float4*)(V + (size_t)c * DIM))[lane];
        float* o = out + (size_t)r * DIM + lane * 4;
        atomicAdd(o + 0, alpha * v.x);
        atomicAdd(o + 1, alpha * v.y);
        atomicAdd(o + 2, alpha * v.z);
        atomicAdd(o + 3, alpha * v.w);
    }
}

// ---------------------------------------------------------------------------
// Launch
// ---------------------------------------------------------------------------
extern "C" void kernel_launch(void* const* d_in, const int* in_sizes, int n_in,
                              void* d_out, int out_size, void* d_ws, size_t ws_size,
                              hipStream_t stream)
{
    const float*     x  = (const float*)d_in[0];
    const long long* ei = (const long long*)d_in[1];   // int64 [2, E]
    const float*     Wq = (const float*)d_in[2];
    const float*     Wk = (const float*)d_in[3];
    const float*     Wv = (const float*)d_in[4];
    float* out = (float*)d_out;

    char* ws = (char*)d_ws;
    const size_t NF = (size_t)NODES * DIM;             // 1,280,000
    float*    LQ     = (float*)(ws);
    float*    LK     = (float*)(ws + NF * 4);
    float*    Vo     = (float*)(ws + NF * 8);
    float*    scores = (float*)(ws + NF * 12);
    unsigned* maxkey = (unsigned*)(ws + NF * 12 + (size_t)EDGES * 4);
    float*    den    = (float*)(ws + NF * 12 + (size_t)EDGES * 4 + (size_t)NODES * 4);

    // 1) QKV projection (fp32 WMMA), leaky fused for Q/K.
    {
        const int waves = 3 * (NODES / 16) * (DIM / 16); // 15000
        const int blk = 128;                              // 4 waves
        const int grid = (waves + 3) / 4;                 // 3750
        qkv_wmma_kernel<<<grid, blk, 0, stream>>>(x, Wq, Wk, Wv, LQ, LK, Vo);
    }
    // 2) init out/den/maxkey
    init_kernel<<<(NODES * DIM + 255) / 256, 256, 0, stream>>>(out, den, maxkey);
    // 3) scores + segment max (wave per edge)
    edge_scores_kernel<<<EDGES / 8, 256, 0, stream>>>(LQ, LK, ei, scores, maxkey);
    // 4) exp + segment sum
    edge_exp_kernel<<<(EDGES + 255) / 256, 256, 0, stream>>>(ei, scores, maxkey, den);
    // 5) weighted scatter-accumulate
    edge_accum_kernel<<<EDGES / 8, 256, 0, stream>>>(ei, scores, den, Vo, out);
}